// SpatialEncoder_56092272886104
// MI455X (gfx1250) — compile-verified
//
#include <hip/hip_runtime.h>

#define NN   100000
#define NE   1000000
#define DH   64
#define LNEPS 1e-5f

typedef float v2f __attribute__((ext_vector_type(2)));
typedef float v8f __attribute__((ext_vector_type(8)));

// ---------------------------------------------------------------- degree prep
__global__ void deg_init_kernel(float* __restrict__ deg) {
    int n = blockIdx.x * blockDim.x + threadIdx.x;
    if (n < NN) deg[n] = 1.0f;                 // self-loop weight
}

__global__ void deg_edge_kernel(const long long* __restrict__ dst,
                                const float* __restrict__ w,
                                float* __restrict__ deg) {
    int e = blockIdx.x * blockDim.x + threadIdx.x;
    if (e < NE) unsafeAtomicAdd(&deg[dst[e]], w[e]);
}

__global__ void dinv_kernel(const float* __restrict__ deg,
                            float* __restrict__ dinv) {
    int n = blockIdx.x * blockDim.x + threadIdx.x;
    if (n < NN) {
        float d = deg[n];
        dinv[n] = (d > 0.0f) ? rsqrtf(d) : 0.0f;
    }
}

// ---------------------------------------------------------------- WMMA GEMM
// out[nrows x 64] = A[nrows x 64] @ W[64 x 64], fp32 via V_WMMA_F32_16X16X4_F32.
// One wave computes one 16-row strip (4 tiles of 16x16 across N). 8 waves/block.
__global__ __launch_bounds__(256) void gemm64_wmma_kernel(
        const float* __restrict__ A, const float* __restrict__ W,
        float* __restrict__ out, int nrows) {
    int wave  = threadIdx.x >> 5;
    int lane  = threadIdx.x & 31;
    int mtile = blockIdx.x * 8 + wave;          // wave-uniform guard (EXEC all-1s)
    if (mtile * 16 >= nrows) return;
    int m0 = mtile * 16;
    int lm = lane & 15;                          // M (A) / N (B,C,D) position
    int lh = lane >> 4;                          // K half-select

    v8f c0 = {}, c1 = {}, c2 = {}, c3 = {};
    const float* arow = A + (size_t)(m0 + lm) * DH;

#pragma unroll
    for (int k0 = 0; k0 < DH; k0 += 4) {
        int ka = k0 + 2 * lh;                    // this lane's K pair base
        v2f a;
        a.x = arow[ka];
        a.y = arow[ka + 1];
        const float* w0 = W + (size_t)ka * DH + lm;   // row ka
        const float* w1 = w0 + DH;                    // row ka+1
        v2f b;
        b.x = w0[0];  b.y = w1[0];
        c0 = __builtin_amdgcn_wmma_f32_16x16x4_f32(false, a, false, b, (short)0, c0, false, false);
        b.x = w0[16]; b.y = w1[16];
        c1 = __builtin_amdgcn_wmma_f32_16x16x4_f32(false, a, false, b, (short)0, c1, false, false);
        b.x = w0[32]; b.y = w1[32];
        c2 = __builtin_amdgcn_wmma_f32_16x16x4_f32(false, a, false, b, (short)0, c2, false, false);
        b.x = w0[48]; b.y = w1[48];
        c3 = __builtin_amdgcn_wmma_f32_16x16x4_f32(false, a, false, b, (short)0, c3, false, false);
    }

    // C/D layout: VGPR r -> M = r + 8*lh, N = lm
    float* orow = out + (size_t)(m0 + 8 * lh) * DH + lm;
#pragma unroll
    for (int r = 0; r < 8; ++r) {
        orow[(size_t)r * DH + 0]  = c0[r];
        orow[(size_t)r * DH + 16] = c1[r];
        orow[(size_t)r * DH + 32] = c2[r];
        orow[(size_t)r * DH + 48] = c3[r];
    }
}

// ---------------------------------------------------------------- aggregation
// agg[n,d] = dinv[n]^2 * xw[n,d] + bias[d]   (self-loop contribution + bias)
__global__ void agg_init_kernel(const float* __restrict__ xw,
                                const float* __restrict__ dinv,
                                const float* __restrict__ bias,
                                float* __restrict__ agg) {
    int i = blockIdx.x * blockDim.x + threadIdx.x;
    if (i < NN * DH) {
        int n = i >> 6, d = i & 63;
        float di = dinv[n];
        agg[i] = di * di * xw[i] + bias[d];
    }
}

// One wave per edge; lane 0 computes coef, broadcast; 2 features per lane.
__global__ __launch_bounds__(256) void agg_edge_kernel(
        const long long* __restrict__ src, const long long* __restrict__ dst,
        const float* __restrict__ w, const float* __restrict__ dinv,
        const float* __restrict__ xw, float* __restrict__ agg) {
    int e    = blockIdx.x * 8 + (threadIdx.x >> 5);
    int lane = threadIdx.x & 31;
    if (e >= NE) return;                          // wave-uniform
    int s = 0, t = 0;
    float coef = 0.0f;
    if (lane == 0) {
        s = (int)src[e];
        t = (int)dst[e];
        coef = dinv[s] * w[e] * dinv[t];
    }
    s    = __shfl(s, 0, 32);
    t    = __shfl(t, 0, 32);
    coef = __shfl(coef, 0, 32);

    float2 v = ((const float2*)(xw + (size_t)s * DH))[lane];
    float* ar = agg + (size_t)t * DH + 2 * lane;
    unsafeAtomicAdd(ar,     coef * v.x);
    unsafeAtomicAdd(ar + 1, coef * v.y);
}

// ---------------------------------------------------------------- LN + ReLU (+res)
// One wave per node (wave32, 2 features/lane), shfl_xor reductions.
__global__ __launch_bounds__(256) void ln_relu_kernel(
        const float* __restrict__ agg, const float* __restrict__ g,
        const float* __restrict__ be, const float* __restrict__ res,
        float* __restrict__ out) {
    int n    = blockIdx.x * 8 + (threadIdx.x >> 5);
    int lane = threadIdx.x & 31;
    if (n >= NN) return;                          // wave-uniform

    float2 v = ((const float2*)(agg + (size_t)n * DH))[lane];
    float sum = v.x + v.y;
#pragma unroll
    for (int off = 16; off; off >>= 1) sum += __shfl_xor(sum, off, 32);
    float mu = sum * (1.0f / 64.0f);
    float dx = v.x - mu, dy = v.y - mu;
    float ss = dx * dx + dy * dy;
#pragma unroll
    for (int off = 16; off; off >>= 1) ss += __shfl_xor(ss, off, 32);
    float rstd = rsqrtf(ss * (1.0f / 64.0f) + LNEPS);

    float2 gg = ((const float2*)g)[lane];
    float2 bb = ((const float2*)be)[lane];
    float y0 = fmaxf(dx * rstd * gg.x + bb.x, 0.0f);
    float y1 = fmaxf(dy * rstd * gg.y + bb.y, 0.0f);
    if (res) {
        float2 r = ((const float2*)(res + (size_t)n * DH))[lane];
        y0 += r.x;
        y1 += r.y;
    }
    float* orow = out + (size_t)n * DH + 2 * lane;
    orow[0] = y0;
    orow[1] = y1;
}

// ---------------------------------------------------------------- launcher
extern "C" void kernel_launch(void* const* d_in, const int* in_sizes, int n_in,
                              void* d_out, int out_size, void* d_ws, size_t ws_size,
                              hipStream_t stream) {
    const float*     x    = (const float*)d_in[0];
    const long long* ei   = (const long long*)d_in[1];   // [2, E] int64
    const float*     ew   = (const float*)d_in[2];
    const float*     W1   = (const float*)d_in[3];
    const float*     b1   = (const float*)d_in[4];
    const float*     g1   = (const float*)d_in[5];
    const float*     be1  = (const float*)d_in[6];
    const float*     W2   = (const float*)d_in[7];
    const float*     b2   = (const float*)d_in[8];
    const float*     g2   = (const float*)d_in[9];
    const float*     be2  = (const float*)d_in[10];
    float*           out  = (float*)d_out;

    const long long* src = ei;
    const long long* dst = ei + NE;

    // workspace partition (floats): deg | dinv | xw | agg | res
    float* ws   = (float*)d_ws;
    float* deg  = ws;
    float* dinv = ws + NN;
    float* xw   = ws + 2 * NN;
    float* agg  = xw + (size_t)NN * DH;
    float* res  = agg + (size_t)NN * DH;

    const int TB = 256;
    const int gN    = (NN + TB - 1) / TB;            // node-wise
    const int gE    = (NE + TB - 1) / TB;            // edge-wise
    const int gND   = (NN * DH + TB - 1) / TB;       // node*feature-wise
    const int gGemm = ((NN / 16) + 7) / 8;           // 6250 mtiles, 8 waves/block
    const int gEdge = (NE + 7) / 8;                  // 8 edges/block
    const int gLN   = (NN + 7) / 8;                  // 8 nodes/block

    // normalization coefficients (shared by both layers)
    deg_init_kernel<<<gN, TB, 0, stream>>>(deg);
    deg_edge_kernel<<<gE, TB, 0, stream>>>(dst, ew, deg);
    dinv_kernel<<<gN, TB, 0, stream>>>(deg, dinv);

    // ---- layer 1
    gemm64_wmma_kernel<<<gGemm, TB, 0, stream>>>(x, W1, xw, NN);
    agg_init_kernel<<<gND, TB, 0, stream>>>(xw, dinv, b1, agg);
    agg_edge_kernel<<<gEdge, TB, 0, stream>>>(src, dst, ew, dinv, xw, agg);
    ln_relu_kernel<<<gLN, TB, 0, stream>>>(agg, g1, be1, nullptr, res);

    // ---- layer 2 (+ residual)
    gemm64_wmma_kernel<<<gGemm, TB, 0, stream>>>(res, W2, xw, NN);
    agg_init_kernel<<<gND, TB, 0, stream>>>(xw, dinv, b2, agg);
    agg_edge_kernel<<<gEdge, TB, 0, stream>>>(src, dst, ew, dinv, xw, agg);
    ln_relu_kernel<<<gLN, TB, 0, stream>>>(agg, g2, be2, res, out);
}